// Radon_515396076141
// MI455X (gfx1250) — compile-verified
//
#include <hip/hip_runtime.h>
#include <math.h>

// Radon forward projection, MI455X (gfx1250, wave32).
// - whole 256x256 fp32 image lives in LDS (256KB dynamic) per workgroup
// - 16 rays per wave; i-axis reduction via V_WMMA_F32_16X16X4_F32 with an
//   all-ones B matrix (D[m][n] = sum_k A[m][k] for every n -> layout-proof)
// - branchless bilinear taps: LDS OOB loads return 0 (hardware zero-padding
//   in y); x-validity folded into the weights with cndmask.

typedef __attribute__((ext_vector_type(2))) float v2f;
typedef __attribute__((ext_vector_type(8))) float v8f;

#define S_PAD 363          // padded size = ceil(sqrt(2)*256)
#define NTH   180          // number of angles
#define IMG_W 256
#define PADB  53           // pad_before
#define RAYS_PER_BATCH (NTH * S_PAD)          // 65340
#define THREADS 512                           // 16 wave32 waves
#define RAYS_PER_BLOCK (16 * (THREADS / 32))  // 256 rays (16 per wave)
#define BLOCKS_PER_BATCH ((RAYS_PER_BATCH + RAYS_PER_BLOCK - 1) / RAYS_PER_BLOCK) // 256

// Branchless bilinear sample of the padded image at (px, py) in padded coords.
// Taps with y outside [0,255] hit LDS addresses outside the 256KB allocation
// -> hardware returns 0 (CDNA5 ISA 4.3.4), matching the zero padding.
// Taps with x outside [0,255] would wrap into the neighboring row, so their
// weight is zeroed instead (separable mask).
__device__ __forceinline__ float bsample(const float* __restrict__ simg,
                                         float px, float py) {
    const float fx = floorf(px), fy = floorf(py);
    const float wx1 = px - fx,  wy1 = py - fy;
    const float wx0 = 1.0f - wx1, wy0 = 1.0f - wy1;
    const int x0 = (int)fx - PADB;
    const int y0 = (int)fy - PADB;

    const int base = y0 * IMG_W + x0;     // may be wildly OOB: loads return 0
    const float v00 = simg[base];
    const float v01 = simg[base + 1];
    const float v10 = simg[base + IMG_W];
    const float v11 = simg[base + IMG_W + 1];

    const float wxm0 = ((unsigned)x0       < (unsigned)IMG_W) ? wx0 : 0.0f;
    const float wxm1 = ((unsigned)(x0 + 1) < (unsigned)IMG_W) ? wx1 : 0.0f;

    return fmaf(wy1, fmaf(wxm1, v11, wxm0 * v10),
                wy0 * fmaf(wxm1, v01, wxm0 * v00));
}

__global__ void radon_wmma_kernel(const float* __restrict__ img,
                                  float* __restrict__ out) {
    extern __shared__ float simg[];            // 256*256 fp32 = 256 KB

    const int b   = blockIdx.x / BLOCKS_PER_BATCH;
    const int blk = blockIdx.x % BLOCKS_PER_BATCH;

    // ---- cooperative load of this batch's image into LDS (float4 stripes) ----
    {
        const float4* __restrict__ src = (const float4*)(img + (size_t)b * IMG_W * IMG_W);
        float4* dst = (float4*)simg;
        for (int l = threadIdx.x; l < (IMG_W * IMG_W / 4); l += THREADS)
            dst[l] = src[l];
    }
    __syncthreads();

    const int lane = threadIdx.x & 31;
    const int wave = threadIdx.x >> 5;
    const int r    = lane & 15;            // ray slot (A-matrix row M)
    const int so   = (lane >> 4) << 1;     // K sub-offset: lanes 0-15 -> 0, 16-31 -> 2

    const int raybase = blk * RAYS_PER_BLOCK + wave * 16;
    const int ray     = raybase + r;
    const bool valid  = ray < RAYS_PER_BATCH;

    const int t = ray / S_PAD;             // angle index
    const int j = ray - t * S_PAD;         // detector column

    // theta_t = t * (180/179) deg = t * pi/179 rad
    const float theta = (float)t * (float)(M_PI / 179.0);
    const float cs = cosf(theta);
    const float sn = sinf(theta);
    const float bj = (float)j * (1.0f / 181.0f) - 1.0f;   // base[j] in [-1,1]
    float A0 = (cs * bj + 1.0f) * 181.0f;  // px(i) = A0 + sn*(i-181)
    float B0 = (1.0f - sn * bj) * 181.0f;  // py(i) = B0 + cs*(i-181)
    if (!valid) { A0 = -1.0e9f; B0 = -1.0e9f; }  // x-mask kills all taps -> 0

    // All-ones B matrix: D[m][n] = sum_k A[m][k] + C[m][n] for every column n.
    v2f bmat; bmat.x = 1.0f; bmat.y = 1.0f;
    v8f acc = {};

    // Main loop: 90 chunks of K=4 cover i = 0..359 (no range guard needed).
    float fi = (float)(so - 181);          // (i0 - 181) as exact float
    for (int n = 0; n < 90; ++n) {
        const float px0 = fmaf(sn, fi, A0);
        const float py0 = fmaf(cs, fi, B0);
        v2f a;
        a.x = bsample(simg, px0,      py0);
        a.y = bsample(simg, px0 + sn, py0 + cs);
        acc = __builtin_amdgcn_wmma_f32_16x16x4_f32(
                  false, a, false, bmat, (short)0, acc, false, false);
        fi += 4.0f;
    }
    // Epilogue chunk: i0 = 360 + so; i = 363 (lanes 16-31, 2nd tap) is masked.
    {
        const float px0 = fmaf(sn, fi, A0);
        const float py0 = fmaf(cs, fi, B0);
        v2f a;
        a.x = bsample(simg, px0, py0);
        const float s1 = bsample(simg, px0 + sn, py0 + cs);
        a.y = (so == 0) ? s1 : 0.0f;
        acc = __builtin_amdgcn_wmma_f32_16x16x4_f32(
                  false, a, false, bmat, (short)0, acc, false, false);
    }

    // D layout (16x16 f32): VGPR k, lanes 0-15 -> M=k, lanes 16-31 -> M=8+k.
    // Every column holds the row sum; lane 0 stores rays 0-7, lane 16 rays 8-15.
    if (lane == 0 || lane == 16) {
        const int mofs = (lane == 16) ? 8 : 0;
        #pragma unroll
        for (int k = 0; k < 8; ++k) {
            const int rr = raybase + mofs + k;
            if (rr < RAYS_PER_BATCH) {
                const int tt = rr / S_PAD;
                const int jj = rr - tt * S_PAD;
                // out[b, 0, jj, tt]
                out[((size_t)b * S_PAD + jj) * NTH + tt] = acc[k];
            }
        }
    }
}

extern "C" void kernel_launch(void* const* d_in, const int* in_sizes, int n_in,
                              void* d_out, int out_size, void* d_ws, size_t ws_size,
                              hipStream_t stream) {
    (void)in_sizes; (void)n_in; (void)d_ws; (void)ws_size; (void)out_size;
    const float* img = (const float*)d_in[0];
    float* out = (float*)d_out;

    const size_t lds_bytes = (size_t)IMG_W * IMG_W * sizeof(float); // 256 KB
    // Opt in to >64KB dynamic LDS (CDNA5 WGP has a 320KB LDS pool).
    // Idempotent + deterministic: called on every launch (no static guards).
    hipFuncSetAttribute((const void*)radon_wmma_kernel,
                        hipFuncAttributeMaxDynamicSharedMemorySize,
                        (int)lds_bytes);

    dim3 grid(2 * BLOCKS_PER_BATCH);   // 512 blocks
    dim3 block(THREADS);               // 16 wave32 waves, 16 rays per wave
    radon_wmma_kernel<<<grid, block, lds_bytes, stream>>>(img, out);
}